// ReshapePagedCache_48928267436361
// MI455X (gfx1250) — compile-verified
//
#include <hip/hip_runtime.h>
#include <hip/hip_bf16.h>

// ---------------------------------------------------------------------------
// reshape_and_cache (paged KV-cache scatter), gfx1250.
// Strategy: inverse-map + gather => exactly 1x read + 1x write of the output
// footprint (~1.07 GB total -> ~46us at 23.3 TB/s), instead of copy+scatter
// (~2.1 GB). Data movement uses the CDNA5 async global<->LDS DMA path
// (ASYNCcnt) with 128-bit transfers.
// ---------------------------------------------------------------------------

#define WG_THREADS 256            // 8 wave32 waves
#define WAVE_CHUNK 2048           // bytes per wave per tensor per chunk (4 x 512B)
#define BLOCK_CHUNK 16384         // bytes per block per tensor per chunk (8 waves)

// Async DMA macros: stringified literal offsets avoid "i"-constraint issues.
#define ASYNC_LD_TO_LDS(ldsa, voff, sbase, IMM)                                \
  asm volatile("global_load_async_to_lds_b128 %0, %1, %2 offset:" #IMM         \
               :: "v"(ldsa), "v"(voff), "s"(sbase) : "memory")
#define ASYNC_ST_FROM_LDS(voff, ldsa, sbase, IMM)                              \
  asm volatile("global_store_async_from_lds_b128 %0, %1, %2 offset:" #IMM      \
               :: "v"(voff), "v"(ldsa), "s"(sbase) : "memory")
#define WAIT_ASYNCCNT0() asm volatile("s_wait_asynccnt 0x0" ::: "memory")

__global__ __launch_bounds__(WG_THREADS)
void rpc_init_inv_kernel(int* __restrict__ inv, int n) {
  int i = blockIdx.x * blockDim.x + threadIdx.x;
  if (i < n) inv[i] = -1;
}

// inv[slot] = token that writes this slot. atomicMax => deterministic winner
// (largest token index) even with duplicate slots; matches reference exactly
// for unique slots.
__global__ __launch_bounds__(WG_THREADS)
void rpc_build_inv_kernel(const int* __restrict__ slots, int* __restrict__ inv,
                          int nt, int total_slots) {
  int t = blockIdx.x * blockDim.x + threadIdx.x;
  if (t < nt) {
    int s = slots[t];
    if (s >= 0 && s < total_slots) atomicMax(&inv[s], t);
  }
}

// One block per destination slot. Copies the 16KB k-row and 16KB v-row for
// this slot (from k/v[token] if a token maps here, else from the old cache)
// through LDS using async b128 DMA ops.
__global__ __launch_bounds__(WG_THREADS)
void rpc_gather_rows_kernel(const float* __restrict__ k,
                            const float* __restrict__ v,
                            const float* __restrict__ kc,
                            const float* __restrict__ vc,
                            const int* __restrict__ inv,
                            float* __restrict__ ok,
                            float* __restrict__ ov,
                            unsigned rowElems /* H*D */) {
  __shared__ __align__(16) unsigned char lds[(WG_THREADS / 32) * 2 * WAVE_CHUNK]; // 32KB

  const unsigned slot     = blockIdx.x;
  const unsigned rowBytes = rowElems * 4u;
  const size_t   rowOff   = (size_t)slot * (size_t)rowElems;

  const int t = inv[slot];
  const float* srcK = (t >= 0) ? (k + (size_t)t * (size_t)rowElems) : (kc + rowOff);
  const float* srcV = (t >= 0) ? (v + (size_t)t * (size_t)rowElems) : (vc + rowOff);

  const unsigned long long sK = (unsigned long long)srcK;
  const unsigned long long sV = (unsigned long long)srcV;
  const unsigned long long dK = (unsigned long long)(ok + rowOff);
  const unsigned long long dV = (unsigned long long)(ov + rowOff);

  const unsigned wave = threadIdx.x >> 5;
  const unsigned lane = threadIdx.x & 31u;

  // Per-wave private LDS staging: [wave*4096, +2048) for K, [+2048, +4096) for V.
  const unsigned ldsK = ((unsigned)(size_t)&lds[0]) + wave * (2u * WAVE_CHUNK) + lane * 16u;
  const unsigned ldsV = ldsK + WAVE_CHUNK;
  const unsigned goBase = wave * WAVE_CHUNK + lane * 16u; // within-row byte offset

  for (unsigned chunk = 0; chunk < rowBytes; chunk += BLOCK_CHUNK) {
    const unsigned go = goBase + chunk;

    // Stage global -> LDS (8 async b128 ops per wave, ASYNCcnt += 8).
    ASYNC_LD_TO_LDS(ldsK, go, sK, 0);
    ASYNC_LD_TO_LDS(ldsK, go, sK, 512);
    ASYNC_LD_TO_LDS(ldsK, go, sK, 1024);
    ASYNC_LD_TO_LDS(ldsK, go, sK, 1536);
    ASYNC_LD_TO_LDS(ldsV, go, sV, 0);
    ASYNC_LD_TO_LDS(ldsV, go, sV, 512);
    ASYNC_LD_TO_LDS(ldsV, go, sV, 1024);
    ASYNC_LD_TO_LDS(ldsV, go, sV, 1536);
    WAIT_ASYNCCNT0();                // loads landed in LDS

    // Drain LDS -> global destination rows.
    ASYNC_ST_FROM_LDS(go, ldsK, dK, 0);
    ASYNC_ST_FROM_LDS(go, ldsK, dK, 512);
    ASYNC_ST_FROM_LDS(go, ldsK, dK, 1024);
    ASYNC_ST_FROM_LDS(go, ldsK, dK, 1536);
    ASYNC_ST_FROM_LDS(go, ldsV, dV, 0);
    ASYNC_ST_FROM_LDS(go, ldsV, dV, 512);
    ASYNC_ST_FROM_LDS(go, ldsV, dV, 1024);
    ASYNC_ST_FROM_LDS(go, ldsV, dV, 1536);
    WAIT_ASYNCCNT0();                // LDS safe to reuse next chunk
  }
}

extern "C" void kernel_launch(void* const* d_in, const int* in_sizes, int n_in,
                              void* d_out, int out_size, void* d_ws, size_t ws_size,
                              hipStream_t stream) {
  const float* k     = (const float*)d_in[0];
  const float* v     = (const float*)d_in[1];
  const float* kc    = (const float*)d_in[2];
  const float* vc    = (const float*)d_in[3];
  const int*   slots = (const int*)d_in[4];   // int32 per harness convention

  const int NT         = in_sizes[4];                 // 16384 tokens
  const int HD         = in_sizes[0] / NT;            // 4096 (= H*D)
  const int totalSlots = in_sizes[2] / HD;            // 16384 (= NB*BS)

  float* ok = (float*)d_out;
  float* ov = ok + (size_t)in_sizes[2];

  int* inv = (int*)d_ws;  // totalSlots ints (64KB)

  rpc_init_inv_kernel<<<(totalSlots + WG_THREADS - 1) / WG_THREADS, WG_THREADS, 0, stream>>>(
      inv, totalSlots);
  rpc_build_inv_kernel<<<(NT + WG_THREADS - 1) / WG_THREADS, WG_THREADS, 0, stream>>>(
      slots, inv, NT, totalSlots);
  rpc_gather_rows_kernel<<<totalSlots, WG_THREADS, 0, stream>>>(
      k, v, kc, vc, inv, ok, ov, (unsigned)HD);
}